// MCL_MAE_58944131170657
// MI455X (gfx1250) — compile-verified
//
#include <hip/hip_runtime.h>
#include <hip/hip_bf16.h>

typedef __attribute__((ext_vector_type(2))) float v2f;
typedef __attribute__((ext_vector_type(8))) float v8f;

static constexpr int Bn = 32768;   // batch
static constexpr int Cn = 1000;    // classes
static constexpr int Kn = 5;       // complementary labels per row
static constexpr int C4 = Cn / 4;  // 250 float4 per row (exact)

__device__ __forceinline__ float wave_max32(float x) {
#pragma unroll
    for (int off = 16; off > 0; off >>= 1)
        x = fmaxf(x, __shfl_xor(x, off, 32));
    return x;
}

// Full 32-lane sum using V_WMMA_F32_16X16X4_F32.
// A-tile (16x4 f32): VGPR0 lanes0-15 -> A[m][0], lanes16-31 -> A[m][2]; VGPR1 = 0.
// B = ones(4x16). D[m][n] = p[m] + p[m+16].
// D layout: acc[r] holds D[r][n] in lanes 0-15 and D[8+r][n] in lanes 16-31.
// Sum of the 8 acc regs gives sum_{m=0..7}(p[m]+p[m+16]) in lanes 0-15 and
// sum_{m=8..15}(...) in lanes 16-31; one xor-16 shuffle completes the total.
__device__ __forceinline__ float wave_sum_wmma(float p) {
    v2f a; a[0] = p;    a[1] = 0.0f;
    v2f b; b[0] = 1.0f; b[1] = 1.0f;
    v8f c = {};
    c = __builtin_amdgcn_wmma_f32_16x16x4_f32(
        /*neg_a=*/false, a, /*neg_b=*/false, b,
        /*c_mod=*/(short)0, c, /*reuse_a=*/false, /*reuse_b=*/false);
    float t = ((c[0] + c[1]) + (c[2] + c[3])) + ((c[4] + c[5]) + (c[6] + c[7]));
    return t + __shfl_xor(t, 16, 32);
}

// One wave32 per row: single pass, row resident in VGPRs (8 x float4 per lane).
__global__ void __launch_bounds__(256)
row_loss_kernel(const float* __restrict__ x,
                const int* __restrict__ labels,
                float* __restrict__ rowloss) {
    const int wave = (blockIdx.x * blockDim.x + threadIdx.x) >> 5;  // == row
    const int lane = threadIdx.x & 31;
    const float* __restrict__ row = x + (size_t)wave * Cn;

    float4 v[8];
    float m = -__builtin_huge_valf();
#pragma unroll
    for (int j = 0; j < 8; ++j) {
        const int c4 = lane + 32 * j;
        if (c4 < C4) {
            v[j] = reinterpret_cast<const float4*>(row)[c4];  // global_load_b128
            m = fmaxf(m, fmaxf(fmaxf(v[j].x, v[j].y), fmaxf(v[j].z, v[j].w)));
        }
    }
    m = wave_max32(m);

    float p = 0.0f;
#pragma unroll
    for (int j = 0; j < 8; ++j) {
        const int c4 = lane + 32 * j;
        if (c4 < C4) {
            p += __expf(v[j].x - m) + __expf(v[j].y - m) +
                 __expf(v[j].z - m) + __expf(v[j].w - m);
        }
    }
    // Branches reconverged; EXEC is all ones here (wave fully active).
    const float denom = wave_sum_wmma(p);

    if (lane == 0) {
        const int* __restrict__ lab = labels + wave * Kn;
        int l[Kn];
        float s = 0.0f;
#pragma unroll
        for (int j = 0; j < Kn; ++j) {
            l[j] = lab[j];
            bool take = (l[j] != -1);
#pragma unroll
            for (int i = 0; i < Kn; ++i)
                if (i < j) take = take && (l[i] != l[j]);
            if (take) {
                const int cl = min(max(l[j], 0), Cn - 1);
                s += __expf(row[cl] - m);
            }
        }
        rowloss[wave] = s / denom;
    }
}

// Deterministic final reduction: fixed-order grid-stride + WMMA wave sums.
__global__ void __launch_bounds__(256)
mean_reduce_kernel(const float* __restrict__ rowloss, float* __restrict__ out) {
    __shared__ float partials[8];
    const int t = threadIdx.x;
    float p = 0.0f;
    for (int i = t; i < Bn; i += 256) p += rowloss[i];
    const float w = wave_sum_wmma(p);
    const int waveId = t >> 5, lane = t & 31;
    if (lane == 0) partials[waveId] = w;
    __syncthreads();
    if (t == 0) {
        float s = 0.0f;
#pragma unroll
        for (int i = 0; i < 8; ++i) s += partials[i];
        out[0] = s * (1.0f / (float)Bn);
    }
}

extern "C" void kernel_launch(void* const* d_in, const int* in_sizes, int n_in,
                              void* d_out, int out_size, void* d_ws, size_t ws_size,
                              hipStream_t stream) {
    const float* x      = (const float*)d_in[0];   // [B, C] float32
    const int*   labels = (const int*)d_in[1];     // [B, K] int
    float* rowloss = (float*)d_ws;                 // Bn floats scratch
    float* out     = (float*)d_out;                // scalar

    // 8 waves per 256-thread block; one wave per row -> Bn/8 blocks.
    row_loss_kernel<<<Bn / 8, 256, 0, stream>>>(x, labels, rowloss);
    mean_reduce_kernel<<<1, 256, 0, stream>>>(rowloss, out);
}